// SARep_5231270166862
// MI455X (gfx1250) — compile-verified
//
#include <hip/hip_runtime.h>
#include <hip/hip_bf16.h>

// Problem constants (match reference)
#define B_    4
#define N_    8192
#define CIN_  64
#define COUT_ 128
#define MID_  64
#define NS_   32
#define S_    2048
#define SK_   (S_ * NS_)      // 65536 grouped columns per batch
#define RADIUS_ 0.1f
#define R2_     (RADIUS_ * RADIUS_)
#define EPS_    1e-5f

typedef float v2f __attribute__((ext_vector_type(2)));
typedef float v8f __attribute__((ext_vector_type(8)));

// Async global->LDS copy of one 16-byte chunk (CDNA5 GLOBAL_LOAD_ASYNC_TO_LDS,
// ASYNCcnt-tracked, bypasses VGPRs). lds = low 32 bits of generic LDS pointer
// (ISA: flat->LDS mapping uses addr[31:0]).
__device__ __forceinline__ void async_copy_b128(unsigned lds,
                                                unsigned long long gaddr) {
  asm volatile("global_load_async_to_lds_b128 %0, %1, off"
               :
               : "v"(lds), "v"(gaddr)
               : "memory");
}
__device__ __forceinline__ void async_wait0() {
  asm volatile("s_wait_asynccnt 0x0" ::: "memory");
}

// ---------------------------------------------------------------------------
// 1) Furthest point sampling: one block per batch, dists held in registers,
//    wave shfl_xor argmax + LDS cross-wave reduce. Writes idx and new_p.
// ---------------------------------------------------------------------------
__global__ __launch_bounds__(1024) void fps_kernel(const float* __restrict__ p,
                                                   int* __restrict__ idx,
                                                   float* __restrict__ newp) {
  const int b = blockIdx.x;
  const int tid = threadIdx.x;
  const int lane = tid & 31, wv = tid >> 5;
  const float* pb = p + (size_t)b * N_ * 3;
  float dist[8];
#pragma unroll
  for (int j = 0; j < 8; ++j) dist[j] = 1e10f;
  __shared__ float swv[32];
  __shared__ int   swi[32];
  __shared__ int   slast;
  int last = 0;
  for (int it = 0; it < S_; ++it) {
    if (tid == 0) {
      idx[b * S_ + it] = last;
      newp[((size_t)b * S_ + it) * 3 + 0] = pb[last * 3 + 0];
      newp[((size_t)b * S_ + it) * 3 + 1] = pb[last * 3 + 1];
      newp[((size_t)b * S_ + it) * 3 + 2] = pb[last * 3 + 2];
    }
    const float lx = pb[last * 3 + 0];
    const float ly = pb[last * 3 + 1];
    const float lz = pb[last * 3 + 2];
    float best = -1.0f; int bi = 0;
#pragma unroll
    for (int j = 0; j < 8; ++j) {
      const int n = j * 1024 + tid;
      const float dx = pb[n * 3 + 0] - lx;
      const float dy = pb[n * 3 + 1] - ly;
      const float dz = pb[n * 3 + 2] - lz;
      const float d = dx * dx + dy * dy + dz * dz;
      dist[j] = fminf(dist[j], d);
      if (dist[j] > best) { best = dist[j]; bi = n; }
    }
#pragma unroll
    for (int off = 16; off > 0; off >>= 1) {
      const float ov = __shfl_xor(best, off, 32);
      const int   oi = __shfl_xor(bi, off, 32);
      if (ov > best) { best = ov; bi = oi; }
    }
    if (lane == 0) { swv[wv] = best; swi[wv] = bi; }
    __syncthreads();
    if (tid == 0) {
      float bv = swv[0]; int ii = swi[0];
      for (int w = 1; w < 32; ++w)
        if (swv[w] > bv) { bv = swv[w]; ii = swi[w]; }
      slast = ii;
    }
    __syncthreads();
    last = slast;
  }
}

// ---------------------------------------------------------------------------
// 2) WMMA GEMM: OUT[b][m][n] = sum_k W[m][k] * act(scale[k]*SRC[b][k][n]+shift[k])
//    K == 64. Block = 128 threads = 4 waves, each wave one 16-row M tile
//    (block covers 64 rows, grid.y = M/64), one 16-col N tile per block.
//    Tiles are DMA'd global->LDS with GLOBAL_LOAD_ASYNC_TO_LDS_B128. For the
//    ACT variant, BN scale/shift + relu is applied IN PLACE in LDS after the
//    DMA completes, so the WMMA K-loop is branch-free in both variants.
// ---------------------------------------------------------------------------
template <int ACT>
__global__ __launch_bounds__(128) void gemm_bn_wmma(
    const float* __restrict__ W, const float* __restrict__ SRC,
    float* __restrict__ OUT, const float* __restrict__ scale,
    const float* __restrict__ shift, int M, int K, int Ntot) {
  const int b = blockIdx.z;
  const int tid = threadIdx.x;
  const int lane = tid & 31, wv = tid >> 5;
  const int m0 = blockIdx.y * 64;
  const int n0 = blockIdx.x * 16;

  __shared__ float sA[64 * 64];   // W rows m0..m0+63, all K (raw)
  __shared__ float sB[64 * 16];   // SRC K x 16 tile
  __shared__ float sSc[64], sSh[64];

  // W tile: 16KB contiguous (rows m0..m0+63, row-major MxK).
  {
    const unsigned long long gW =
        (unsigned long long)(const void*)(W + (size_t)m0 * K);
    for (int i = tid; i < (64 * 64) / 4; i += 128) {
      const unsigned lds = (unsigned)(size_t)(const void*)&sA[i * 4];
      async_copy_b128(lds, gW + (unsigned long long)i * 16ull);
    }
    // SRC tile: K rows of 16 floats (64B each), 16B chunks.
    for (int i = tid; i < (K * 16) / 4; i += 128) {
      const int k = i >> 2, c4 = (i & 3) * 4;
      const unsigned lds = (unsigned)(size_t)(const void*)&sB[k * 16 + c4];
      const unsigned long long ga = (unsigned long long)(const void*)
          &SRC[((size_t)b * K + k) * Ntot + n0 + c4];
      async_copy_b128(lds, ga);
    }
  }
  if (ACT && tid < 64) { sSc[tid] = scale[tid]; sSh[tid] = shift[tid]; }
  async_wait0();
  __syncthreads();
  if (ACT) {
    // In-place BN + relu on the staged activation tile (8 elems/thread).
    for (int i = tid; i < K * 16; i += 128) {
      const int k = i >> 4;
      sB[i] = fmaxf(fmaf(sSc[k], sB[i], sSh[k]), 0.0f);
    }
    __syncthreads();
  }

  const int mr = wv * 16 + (lane & 15);
  const int khalf = (lane >> 4) * 2;   // 0 or 2 (A K-pair select)
  const int klo = (lane >> 4);         // 0 or 1 (B row select)
  const int col16 = lane & 15;
  v8f acc = {0.f, 0.f, 0.f, 0.f, 0.f, 0.f, 0.f, 0.f};
#pragma unroll
  for (int k0 = 0; k0 < 64; k0 += 4) {
    v2f a, bb;
    a.x = sA[mr * 64 + k0 + khalf + 0];
    a.y = sA[mr * 64 + k0 + khalf + 1];
    bb.x = sB[(k0 + klo) * 16 + col16];
    bb.y = sB[(k0 + 2 + klo) * 16 + col16];
    acc = __builtin_amdgcn_wmma_f32_16x16x4_f32(false, a, false, bb,
                                                (short)0, acc, false, false);
  }
  const int col = n0 + col16;
  const int rowBase = m0 + wv * 16 + ((lane >> 4) << 3);
#pragma unroll
  for (int r = 0; r < 8; ++r)
    OUT[((size_t)b * M + rowBase + r) * Ntot + col] = acc[r];
}

// ---------------------------------------------------------------------------
// 3) Ball query (first-K-in-radius): one thread per center.
// ---------------------------------------------------------------------------
__global__ __launch_bounds__(256) void ball_query_kernel(
    const float* __restrict__ newp, const float* __restrict__ p,
    int* __restrict__ gidx) {
  const int t = blockIdx.x * blockDim.x + threadIdx.x;
  if (t >= B_ * S_) return;
  const int b = t / S_;
  const float cx = newp[(size_t)t * 3 + 0];
  const float cy = newp[(size_t)t * 3 + 1];
  const float cz = newp[(size_t)t * 3 + 2];
  const float* pb = p + (size_t)b * N_ * 3;
  int* gp = gidx + (size_t)t * NS_;
  int cnt = 0, first = 0;
  for (int n = 0; n < N_; ++n) {
    const float dx = pb[n * 3 + 0] - cx;
    const float dy = pb[n * 3 + 1] - cy;
    const float dz = pb[n * 3 + 2] - cz;
    if (dx * dx + dy * dy + dz * dz < R2_) {
      if (cnt == 0) first = n;
      gp[cnt++] = n;
      if (cnt >= NS_) break;
    }
  }
  for (int k = cnt; k < NS_; ++k) gp[k] = (cnt > 0) ? first : 0;
}

// ---------------------------------------------------------------------------
// 4) Positional-encoding GEMM: t3 = w2a(64x3) @ dp, dp built in LDS from the
//    gathered/centered/scaled coordinates. K padded 3->4, single WMMA/tile.
// ---------------------------------------------------------------------------
__global__ __launch_bounds__(128) void posenc_gemm_wmma(
    const float* __restrict__ w2a, const float* __restrict__ p,
    const float* __restrict__ newp, const int* __restrict__ gidx,
    float* __restrict__ t3) {
  const int b = blockIdx.z;
  const int tid = threadIdx.x;
  const int lane = tid & 31, wv = tid >> 5;
  const int n0 = blockIdx.x * 16;
  __shared__ float sB[4 * 16];
  if (tid < 16) {
    const int nflat = n0 + tid;
    const int s = nflat >> 5, kk = nflat & 31;
    const int g = gidx[((size_t)b * S_ + s) * NS_ + kk];
    const float* pp = p + ((size_t)b * N_ + g) * 3;
    const float* cp = newp + ((size_t)b * S_ + s) * 3;
    const float inv = 1.0f / RADIUS_;
    sB[0 * 16 + tid] = (pp[0] - cp[0]) * inv;
    sB[1 * 16 + tid] = (pp[1] - cp[1]) * inv;
    sB[2 * 16 + tid] = (pp[2] - cp[2]) * inv;
    sB[3 * 16 + tid] = 0.0f;
  }
  __syncthreads();
  const int m = wv * 16 + (lane & 15);
  const int khalf = (lane >> 4) * 2;
  v2f a, bb;
  a.x = w2a[m * 3 + khalf];                              // k = 0 or 2
  a.y = (khalf + 1 < 3) ? w2a[m * 3 + khalf + 1] : 0.0f; // k = 1 or pad
  bb.x = sB[(lane >> 4) * 16 + (lane & 15)];
  bb.y = sB[(2 + (lane >> 4)) * 16 + (lane & 15)];
  v8f acc = {0.f, 0.f, 0.f, 0.f, 0.f, 0.f, 0.f, 0.f};
  acc = __builtin_amdgcn_wmma_f32_16x16x4_f32(false, a, false, bb,
                                              (short)0, acc, false, false);
  const int col = n0 + (lane & 15);
  const int rowBase = wv * 16 + ((lane >> 4) << 3);
#pragma unroll
  for (int r = 0; r < 8; ++r)
    t3[((size_t)b * MID_ + rowBase + r) * SK_ + col] = acc[r];
}

// ---------------------------------------------------------------------------
// 5) Per-channel sum / sum-of-squares reduction (training BN stats).
//    src layout (B, C, per); grid = (chunks, C); block atomicAdd once.
// ---------------------------------------------------------------------------
__global__ __launch_bounds__(256) void stats_kernel(
    const float* __restrict__ src, float* __restrict__ acc, int C,
    int perBC, int B) {
  const int c = blockIdx.y;
  const int tid = threadIdx.x;
  const long long total = (long long)B * perBC;
  float s = 0.f, q = 0.f;
  for (long long i = (long long)blockIdx.x * blockDim.x + tid; i < total;
       i += (long long)gridDim.x * blockDim.x) {
    const long long bb = i / perBC, off = i % perBC;
    const float v = src[(bb * C + c) * perBC + off];
    s += v; q += v * v;
  }
  __shared__ float rs[256], rq[256];
  rs[tid] = s; rq[tid] = q;
  __syncthreads();
  for (int st = 128; st > 0; st >>= 1) {
    if (tid < st) { rs[tid] += rs[tid + st]; rq[tid] += rq[tid + st]; }
    __syncthreads();
  }
  if (tid == 0) {
    atomicAdd(&acc[c], rs[0]);
    atomicAdd(&acc[C + c], rq[0]);
  }
}

__global__ void bn_finalize(const float* __restrict__ acc,
                            const float* __restrict__ g,
                            const float* __restrict__ bta,
                            float* __restrict__ scale, float* __restrict__ shift,
                            int C, float inv_cnt) {
  const int c = threadIdx.x;
  if (c < C) {
    const float mean = acc[c] * inv_cnt;
    const float var = acc[C + c] * inv_cnt - mean * mean;
    const float sc = g[c] * rsqrtf(var + EPS_);
    scale[c] = sc;
    shift[c] = bta[c] - mean * sc;
  }
}

// ---------------------------------------------------------------------------
// 6) Residual branch: identity = wskip @ f[:, :, idx] + bskip
// ---------------------------------------------------------------------------
__global__ __launch_bounds__(256) void skip_kernel(
    const float* __restrict__ wskip, const float* __restrict__ bskip,
    const float* __restrict__ f, const int* __restrict__ idx,
    float* __restrict__ ident) {
  const int t = blockIdx.x * blockDim.x + threadIdx.x;
  if (t >= B_ * COUT_ * S_) return;
  const int s = t % S_;
  const int o = (t / S_) % COUT_;
  const int b = t / (S_ * COUT_);
  const int ii = idx[b * S_ + s];
  const float* fb = f + (size_t)b * CIN_ * N_;
  const float* wr = wskip + (size_t)o * CIN_;
  float a = bskip[o];
#pragma unroll 8
  for (int c = 0; c < CIN_; ++c) a = fmaf(wr[c], fb[(size_t)c * N_ + ii], a);
  ident[t] = a;
}

// ---------------------------------------------------------------------------
// 7) Final fuse: out = relu( max_k( bn2b(t4) + bn1b(t2)[gidx] ) + identity )
// ---------------------------------------------------------------------------
__global__ __launch_bounds__(256) void final_kernel(
    const float* __restrict__ t4, const float* __restrict__ t2,
    const int* __restrict__ gidx, const float* __restrict__ ident,
    const float* __restrict__ sc2, const float* __restrict__ sh2,
    const float* __restrict__ sc4, const float* __restrict__ sh4,
    float* __restrict__ outF) {
  const int t = blockIdx.x * blockDim.x + threadIdx.x;
  if (t >= B_ * COUT_ * S_) return;
  const int s = t % S_;
  const int o = (t / S_) % COUT_;
  const int b = t / (S_ * COUT_);
  const float a2 = sc2[o], c2 = sh2[o];
  const float a4 = sc4[o], c4 = sh4[o];
  const float* t4p = t4 + ((size_t)b * COUT_ + o) * SK_ + (size_t)s * NS_;
  const float* t2p = t2 + ((size_t)b * COUT_ + o) * N_;
  const int* gp = gidx + ((size_t)b * S_ + s) * NS_;
  __builtin_prefetch(t4p + 64, 0, 1);   // stream-ahead hint (global_prefetch_b8)
  float m = -3.402823466e38f;
#pragma unroll 8
  for (int k = 0; k < NS_; ++k) {
    const float pe = fmaf(a4, t4p[k], c4);
    const float fj = fmaf(a2, t2p[gp[k]], c2);
    m = fmaxf(m, pe + fj);
  }
  outF[t] = fmaxf(m + ident[t], 0.0f);
}

// ---------------------------------------------------------------------------
extern "C" void kernel_launch(void* const* d_in, const int* in_sizes, int n_in,
                              void* d_out, int out_size, void* d_ws, size_t ws_size,
                              hipStream_t stream) {
  const float* p     = (const float*)d_in[0];
  const float* f     = (const float*)d_in[1];
  const float* w1a   = (const float*)d_in[2];
  const float* g1a   = (const float*)d_in[3];
  const float* b1a   = (const float*)d_in[4];
  const float* w1b   = (const float*)d_in[5];
  const float* g1b   = (const float*)d_in[6];
  const float* b1b   = (const float*)d_in[7];
  const float* w2a   = (const float*)d_in[8];
  const float* g2a   = (const float*)d_in[9];
  const float* b2a   = (const float*)d_in[10];
  const float* w2b   = (const float*)d_in[11];
  const float* g2b   = (const float*)d_in[12];
  const float* b2b   = (const float*)d_in[13];
  const float* wskip = (const float*)d_in[14];
  const float* bskip = (const float*)d_in[15];

  float* newp = (float*)d_out;                       // (B,S,3) flat = 24576
  float* outF = (float*)d_out + (size_t)B_ * S_ * 3; // (B,COUT,S)

  // Workspace carve-out (~222 MB)
  char* cur = (char*)d_ws;
  auto carve = [&](size_t bytes) -> void* {
    void* r = (void*)cur;
    cur += (bytes + 255) & ~(size_t)255;
    return r;
  };
  float* t1    = (float*)carve(sizeof(float) * (size_t)B_ * MID_ * N_);
  float* t2    = (float*)carve(sizeof(float) * (size_t)B_ * COUT_ * N_);
  float* t3    = (float*)carve(sizeof(float) * (size_t)B_ * MID_ * SK_);
  float* t4    = (float*)carve(sizeof(float) * (size_t)B_ * COUT_ * SK_);
  float* ident = (float*)carve(sizeof(float) * (size_t)B_ * COUT_ * S_);
  float* stats = (float*)carve(sizeof(float) * 4 * 2 * COUT_);
  float* sc    = (float*)carve(sizeof(float) * 8 * COUT_);
  int*   idx   = (int*)carve(sizeof(int) * (size_t)B_ * S_);
  int*   gidx  = (int*)carve(sizeof(int) * (size_t)B_ * S_ * NS_);

  float* st1 = stats + 0 * COUT_;
  float* st2 = stats + 2 * COUT_;
  float* st3 = stats + 4 * COUT_;
  float* st4 = stats + 6 * COUT_;
  float* sc1 = sc + 0 * COUT_; float* sh1 = sc + 1 * COUT_;
  float* sc2 = sc + 2 * COUT_; float* sh2 = sc + 3 * COUT_;
  float* sc3 = sc + 4 * COUT_; float* sh3 = sc + 5 * COUT_;
  float* sc4 = sc + 6 * COUT_; float* sh4 = sc + 7 * COUT_;

  hipMemsetAsync(stats, 0, sizeof(float) * 4 * 2 * COUT_, stream);

  // FPS -> idx, new_p
  fps_kernel<<<B_, 1024, 0, stream>>>(p, idx, newp);

  // convs1: t1 = w1a @ f ; BN stats ; t2 = w1b @ relu(bn(t1)) ; BN stats
  gemm_bn_wmma<0><<<dim3(N_ / 16, MID_ / 64, B_), 128, 0, stream>>>(
      w1a, f, t1, nullptr, nullptr, MID_, CIN_, N_);
  stats_kernel<<<dim3(32, MID_), 256, 0, stream>>>(t1, st1, MID_, N_, B_);
  bn_finalize<<<1, COUT_, 0, stream>>>(st1, g1a, b1a, sc1, sh1, MID_,
                                       1.0f / ((float)B_ * (float)N_));
  gemm_bn_wmma<1><<<dim3(N_ / 16, COUT_ / 64, B_), 128, 0, stream>>>(
      w1b, t1, t2, sc1, sh1, COUT_, MID_, N_);
  stats_kernel<<<dim3(32, COUT_), 256, 0, stream>>>(t2, st2, COUT_, N_, B_);
  bn_finalize<<<1, COUT_, 0, stream>>>(st2, g1b, b1b, sc2, sh2, COUT_,
                                       1.0f / ((float)B_ * (float)N_));

  // grouping
  ball_query_kernel<<<(B_ * S_) / 256, 256, 0, stream>>>(newp, p, gidx);

  // convs2: t3 = w2a @ dp ; BN stats ; t4 = w2b @ relu(bn(t3)) ; BN stats
  posenc_gemm_wmma<<<dim3(SK_ / 16, 1, B_), 128, 0, stream>>>(
      w2a, p, newp, gidx, t3);
  stats_kernel<<<dim3(32, MID_), 256, 0, stream>>>(t3, st3, MID_, SK_, B_);
  bn_finalize<<<1, COUT_, 0, stream>>>(st3, g2a, b2a, sc3, sh3, MID_,
                                       1.0f / ((float)B_ * (float)SK_));
  gemm_bn_wmma<1><<<dim3(SK_ / 16, COUT_ / 64, B_), 128, 0, stream>>>(
      w2b, t3, t4, sc3, sh3, COUT_, MID_, SK_);
  stats_kernel<<<dim3(32, COUT_), 256, 0, stream>>>(t4, st4, COUT_, SK_, B_);
  bn_finalize<<<1, COUT_, 0, stream>>>(st4, g2b, b2b, sc4, sh4, COUT_,
                                       1.0f / ((float)B_ * (float)SK_));

  // residual + final fuse
  skip_kernel<<<(B_ * COUT_ * S_) / 256, 256, 0, stream>>>(wskip, bskip, f,
                                                           idx, ident);
  final_kernel<<<(B_ * COUT_ * S_) / 256, 256, 0, stream>>>(
      t4, t2, gidx, ident, sc2, sh2, sc4, sh4, outF);
}